// BiasBERTModel_3805341024371
// MI455X (gfx1250) — compile-verified
//
#include <hip/hip_runtime.h>
#include <math.h>

#define B_DIM 4
#define S_DIM 128
#define NUM_ITEMS 10000
#define NUM_SPECIAL 3
#define VOCAB_SZ (NUM_ITEMS + NUM_SPECIAL)
#define PAD_ID NUM_ITEMS
#define THREADS 256
#define WAVES (THREADS / 32)
#define TILES (NUM_ITEMS / 16) /* 625, exact */

typedef __attribute__((ext_vector_type(2))) float v2f;
typedef __attribute__((ext_vector_type(8))) float v8f;

// Module-scope scratch: reciprocal pop biases, rebuilt every kernel_launch call
// before use (no cross-call state relied upon).
__device__ float g_inv_pop[VOCAB_SZ];

// Branchless exact-GELU: 0.5*h*(1+erf(h/sqrt2)), Abramowitz-Stegun 7.1.26 erf
// (|abs err| <= 1.5e-7): one v_rcp_f32 + one v_exp_f32 + fma chain, no EXEC branches.
__device__ __forceinline__ float gelu_exact(float h)
{
    const float z  = h * 0.70710678118654752f;
    const float az = fabsf(z);
    const float t  = __builtin_amdgcn_rcpf(fmaf(0.3275911f, az, 1.0f));
    float p = fmaf(t, 1.061405429f, -1.453152027f);
    p = fmaf(t, p, 1.421413741f);
    p = fmaf(t, p, -0.284496736f);
    p = fmaf(t, p, 0.254829592f);
    p = p * t;
    const float e  = __expf(-az * az);
    float er = fmaf(-p, e, 1.0f);
    er = copysignf(er, z);
    return 0.5f * h * (1.0f + er);
}

// One-time precompute: inv_pop[v] = pop[v]!=0 ? 1/pop[v] : 0.
// div_no_nan(x, d) == x * inv_pop (0-denominator maps to 0 via x*0).
__global__ void init_inv_pop_kernel(const float* __restrict__ pop)
{
    const int i = blockIdx.x * blockDim.x + threadIdx.x;
    if (i < VOCAB_SZ) {
        const float pb = pop[i];
        g_inv_pop[i] = (pb != 0.0f) ? (1.0f / pb) : 0.0f;
    }
}

__global__ __launch_bounds__(THREADS)
void bias_bert_ce_kernel(const int* __restrict__ seqs,
                         const int* __restrict__ labels,
                         const float* __restrict__ t_src,
                         const float* __restrict__ t_dst,
                         const float* __restrict__ pop_norm,
                         const float* __restrict__ W1,
                         const float* __restrict__ b1,
                         const float* __restrict__ W2,
                         const float* __restrict__ b2,
                         float* __restrict__ out)
{
    __shared__ float lds_logits[NUM_ITEMS];   // 40 KB of the 320 KB WGP LDS
    __shared__ float lds_red[WAVES];
    __shared__ float lds_scalar;

    const int pos = blockIdx.x;               // 0 .. B*S-1
    const int b   = pos / S_DIM;
    const int s   = pos % S_DIM;

    const int v_prev = (s == 0)         ? PAD_ID : max(seqs[b * S_DIM + (s - 1)], 0);
    const int v_next = (s == S_DIM - 1) ? PAD_ID : max(seqs[b * S_DIM + (s + 1)], 0);
    const float* __restrict__ rs = t_src + (long)v_prev * VOCAB_SZ;
    const float* __restrict__ rd = t_dst + (long)v_next * VOCAB_SZ;

    const int tid  = threadIdx.x;
    const int lane = tid & 31;
    const int wave = tid >> 5;
    const int half = lane >> 4;   // 0: lanes 0-15 carry A K=0,1 ; 1: lanes 16-31 carry K=2,3
    const int l16  = lane & 15;
    const bool lo_half = (half == 0);

    // B operand tiles for layer-1 (W1 is 3x32 row-major), K padded 3->4 with zeros.
    // Branchless build: load both row candidates, select per half.
    {
    }
    const float w1r0_lo = W1[0 * 32 + l16];
    const float w1r1_lo = W1[1 * 32 + l16];
    const float w1r2_lo = W1[2 * 32 + l16];
    const float w1r0_hi = W1[0 * 32 + 16 + l16];
    const float w1r1_hi = W1[1 * 32 + 16 + l16];
    const float w1r2_hi = W1[2 * 32 + 16 + l16];
    v2f Bt0, Bt1;
    Bt0.x = lo_half ? w1r0_lo : w1r2_lo;   Bt0.y = lo_half ? w1r1_lo : 0.0f;
    Bt1.x = lo_half ? w1r0_hi : w1r2_hi;   Bt1.y = lo_half ? w1r1_hi : 0.0f;

    const float c0    = b1[l16];
    const float c1    = b1[16 + l16];
    const float w2lo  = W2[l16];
    const float w2hi  = W2[16 + l16];
    const float bias2 = b2[0];

    float lmax = -3.402823466e38f;

    // Branchless A-operand tile load: every lane loads all streams (lanes 16-31
    // hit the same cache lines as 0-15), then v_cndmask selects per half.
    auto load_tile = [&](int tt) -> v2f {
        const int v = tt * 16 + l16;
        const float inv = g_inv_pop[v];
        const float fs  = rs[v];
        const float fd  = rd[v];
        const float pn  = pop_norm[v];
        v2f a;
        a.x = lo_half ? fs * inv : pn;
        a.y = lo_half ? fd * inv : 0.0f;
        return a;
    };

    // ---- software-pipelined tile loop; wave-uniform trip count -> EXEC all-ones
    // at every WMMA.
    int t = wave;
    v2f A = load_tile(t);

    for (; t < TILES; t += WAVES) {
        const int v0 = t * 16;

        // issue next tile's loads before the GELU chain
        const int tn = t + WAVES;
        const v2f An = (tn < TILES) ? load_tile(tn) : A;

        v8f acc0 = { c0, c0, c0, c0, c0, c0, c0, c0 };
        v8f acc1 = { c1, c1, c1, c1, c1, c1, c1, c1 };
        acc0 = __builtin_amdgcn_wmma_f32_16x16x4_f32(false, A, false, Bt0, (short)0, acc0, false, false);
        acc1 = __builtin_amdgcn_wmma_f32_16x16x4_f32(false, A, false, Bt1, (short)0, acc1, false, false);

        // exact-GELU (branchless) + layer-2 partial for this lane's column
        float part[8];
        #pragma unroll
        for (int r = 0; r < 8; ++r) {
            const float h0 = gelu_exact(acc0[r]);
            const float h1 = gelu_exact(acc1[r]);
            part[r] = fmaf(h0, w2lo, h1 * w2hi);
        }
        // reduce across the 16 lanes of each half (rows r / 8+r of the D tile)
        #pragma unroll
        for (int m = 1; m <= 8; m <<= 1) {
            #pragma unroll
            for (int r = 0; r < 8; ++r)
                part[r] += __shfl_xor(part[r], m, 32);
        }
        #pragma unroll
        for (int r = 0; r < 8; ++r) {
            const float lg = part[r] + bias2;
            if (l16 == 0) lds_logits[v0 + half * 8 + r] = lg;
            lmax = fmaxf(lmax, lg);
        }

        A = An;
    }

    // ---- block max
    #pragma unroll
    for (int m = 1; m < 32; m <<= 1) lmax = fmaxf(lmax, __shfl_xor(lmax, m, 32));
    if (lane == 0) lds_red[wave] = lmax;
    __syncthreads();
    if (tid == 0) {
        float mx = lds_red[0];
        for (int w = 1; w < WAVES; ++w) mx = fmaxf(mx, lds_red[w]);
        lds_scalar = mx;
    }
    __syncthreads();
    const float mx = lds_scalar;

    // ---- exp-sum over LDS logits (branchless fast exp)
    float lsum = 0.0f;
    for (int i = tid; i < NUM_ITEMS; i += THREADS)
        lsum += __expf(lds_logits[i] - mx);
    #pragma unroll
    for (int m = 1; m < 32; m <<= 1) lsum += __shfl_xor(lsum, m, 32);
    if (lane == 0) lds_red[wave] = lsum;
    __syncthreads();

    if (tid == 0) {
        float sm = 0.0f;
        for (int w = 0; w < WAVES; ++w) sm += lds_red[w];
        const int lab = labels[pos];
        float ce = 0.0f;
        if (lab != -100) {
            const int li = max(lab, 0);
            if (li < NUM_ITEMS)
                ce = -(lds_logits[li] - mx - logf(sm));
        }
        out[pos] = ce;
    }
}

extern "C" void kernel_launch(void* const* d_in, const int* in_sizes, int n_in,
                              void* d_out, int out_size, void* d_ws, size_t ws_size,
                              hipStream_t stream)
{
    (void)in_sizes; (void)n_in; (void)out_size; (void)d_ws; (void)ws_size;
    const int*   seqs   = (const int*)d_in[0];
    const int*   labels = (const int*)d_in[1];
    const float* tsrc   = (const float*)d_in[2];
    const float* tdst   = (const float*)d_in[3];
    const float* pop    = (const float*)d_in[4];
    const float* popn   = (const float*)d_in[5];
    const float* W1     = (const float*)d_in[6];
    const float* b1     = (const float*)d_in[7];
    const float* W2     = (const float*)d_in[8];
    const float* b2     = (const float*)d_in[9];
    float* out = (float*)d_out;

    hipLaunchKernelGGL(init_inv_pop_kernel,
                       dim3((VOCAB_SZ + 255) / 256), dim3(256), 0, stream, pop);
    hipLaunchKernelGGL(bias_bert_ce_kernel,
                       dim3(B_DIM * S_DIM), dim3(THREADS), 0, stream,
                       seqs, labels, tsrc, tdst, popn, W1, b1, W2, b2, out);
}